// ChamferLossSelf_21474836480163
// MI455X (gfx1250) — compile-verified
//
#include <hip/hip_runtime.h>

// ---------------------------------------------------------------------------
// Chamfer-style loss for B=4, N=4096, D=3 point clouds on gfx1250 (MI455X).
// dist(i,j) = xx_i + yy_j - 2 * dot(x_i, y_j); the dot term goes through
// V_WMMA_F32_16X16X4_F32 (K padded 3 -> 4 with zeros, full f32 precision).
// Each wave owns TWO 16-column tiles so one A-fragment / xx load feeds two
// WMMAs; the diag-fill path is compiled out for the cross matrices.
// ---------------------------------------------------------------------------

typedef __attribute__((ext_vector_type(2))) float v2f;
typedef __attribute__((ext_vector_type(8))) float v8f;

#define NPTS 4096
#define NBATCH 4
#define DIAG_FILL_F 500000.0f
#define ALPHA_F 1.0f

// ---------------------------------------------------------------------------
// Kernel 1: column-min of a pairwise distance matrix.
//   SELF=false: z=0: X=gts,Y=preds -> minA ; z=1: X=preds,Y=gts -> minB
//   SELF=true : z=0: X=Y=gts      -> mins1; z=1: X=Y=preds     -> mins2
// ---------------------------------------------------------------------------
template <bool SELF>
__global__ __launch_bounds__(256) void chamfer_min_kernel(
    const float* __restrict__ gts, const float* __restrict__ preds,
    float* __restrict__ ws)
{
  extern __shared__ float s[];            // 4 * 4096 floats = 64 KB
  float* sx  = s;
  float* sy  = s + NPTS;
  float* sz  = s + 2 * NPTS;
  float* sxx = s + 3 * NPTS;

  const int zi = blockIdx.z;              // 0 or 1 within this template
  const int b  = blockIdx.y;              // batch
  const int cg = blockIdx.x;              // 256-column group (0..15)

  const float* X;
  const float* Y;
  float* outp;
  if (SELF) {
    X = zi ? preds : gts;
    Y = X;
    outp = ws + (2 + zi) * NBATCH * NPTS;        // mins1 / mins2
  } else {
    X = zi ? preds : gts;
    Y = zi ? gts : preds;
    outp = ws + zi * NBATCH * NPTS;              // minA / minB
  }
  X    += (size_t)b * NPTS * 3;
  Y    += (size_t)b * NPTS * 3;
  outp += b * NPTS;

  // Stage X into LDS as SoA, plus xx = |x|^2.
  for (int i = threadIdx.x; i < NPTS; i += blockDim.x) {
    float x0 = X[i * 3 + 0];
    float x1 = X[i * 3 + 1];
    float x2 = X[i * 3 + 2];
    sx[i] = x0; sy[i] = x1; sz[i] = x2;
    sxx[i] = x0 * x0 + x1 * x1 + x2 * x2;
  }
  __syncthreads();

  const int lane = threadIdx.x & 31;
  const int wave = threadIdx.x >> 5;      // 0..7
  const int half = lane >> 4;             // 0: K=0,1 slice   1: K=2,3 slice
  const int l16  = lane & 15;
  const int j0   = cg * 256 + wave * 32;  // first column of this wave's pair
  const int jA   = j0 + l16;              // lane's column in tile A
  const int jB   = j0 + 16 + l16;         // lane's column in tile B

  // B fragments (4x16, column-major per lane) + yy for both column tiles.
  const float a0 = Y[jA * 3 + 0], a1 = Y[jA * 3 + 1], a2 = Y[jA * 3 + 2];
  const float b0 = Y[jB * 3 + 0], b1 = Y[jB * 3 + 1], b2 = Y[jB * 3 + 2];
  const float yyA = a0 * a0 + a1 * a1 + a2 * a2;
  const float yyB = b0 * b0 + b1 * b1 + b2 * b2;
  v2f bfragA, bfragB;
  bfragA.x = half ? a2 : a0;  bfragA.y = half ? 0.0f : a1;
  bfragB.x = half ? b2 : b0;  bfragB.y = half ? 0.0f : b1;

  const float diagA = DIAG_FILL_F - yyA;  // diagonal candidates (shifted space)
  const float diagB = DIAG_FILL_F - yyB;
  const v8f czero = {0.f, 0.f, 0.f, 0.f, 0.f, 0.f, 0.f, 0.f};

  float mA = 3.0e38f;                     // min of (xx_i - 2*dot)
  float mB = 3.0e38f;

  for (int t = 0; t < NPTS / 16; ++t) {
    // A fragment: rows t*16 .. t*16+15 of X (shared by both WMMAs).
    const int row = t * 16 + l16;
    v2f afrag;
    afrag.x = half ? sz[row] : sx[row];
    afrag.y = half ? 0.0f : sy[row];

    v8f zzA = __builtin_amdgcn_wmma_f32_16x16x4_f32(
        false, afrag, false, bfragA, (short)0, czero, false, false);
    v8f zzB = __builtin_amdgcn_wmma_f32_16x16x4_f32(
        false, afrag, false, bfragB, (short)0, czero, false, false);

    // xx for this lane's 8 rows: M = r + 8*half (C/D layout); shared too.
    const int ibase = t * 16 + half * 8;
    const float4* px = (const float4*)(sxx + ibase);   // 32B aligned
    const float4 xa = px[0];
    const float4 xb = px[1];
    const float xr[8] = {xa.x, xa.y, xa.z, xa.w, xb.x, xb.y, xb.z, xb.w};

#pragma unroll
    for (int r = 0; r < 8; ++r) {
      float cA = fmaf(zzA[r], -2.0f, xr[r]);           // xx_i - 2*dot
      float cB = fmaf(zzB[r], -2.0f, xr[r]);
      if (SELF) {
        const int i = ibase + r;
        if (i == jA) cA = diagA;
        if (i == jB) cB = diagB;
      }
      mA = fminf(mA, cA);
      mB = fminf(mB, cB);
    }
  }

  // Combine the two lane halves (same column N, rows 0..7 vs 8..15 mod 16).
  mA = fminf(mA, __shfl_xor(mA, 16, 32));
  mB = fminf(mB, __shfl_xor(mB, 16, 32));
  if (half == 0) {
    outp[jA] = mA + yyA;
    outp[jB] = mB + yyB;
  }
}

// ---------------------------------------------------------------------------
// Kernel 2: per batch -> bitonic-sort mins1 & mins2 (together), ordering sum,
// cross-min sums, final scalar.
// ---------------------------------------------------------------------------
__global__ __launch_bounds__(1024) void chamfer_finalize_kernel(
    const float* __restrict__ ws, float* __restrict__ out)
{
  __shared__ float sA[NPTS];
  __shared__ float sB[NPTS];

  const int b   = blockIdx.x;
  const int tid = threadIdx.x;
  const int nt  = blockDim.x;

  const float* minA = ws + 0 * NBATCH * NPTS + b * NPTS;
  const float* minB = ws + 1 * NBATCH * NPTS + b * NPTS;
  const float* m1   = ws + 2 * NBATCH * NPTS + b * NPTS;
  const float* m2   = ws + 3 * NBATCH * NPTS + b * NPTS;

  float crossAcc = 0.0f;
  for (int i = tid; i < NPTS; i += nt) {
    sA[i] = m1[i];
    sB[i] = m2[i];
    crossAcc += minA[i] + minB[i];
  }
  __syncthreads();

  // Bitonic sort (ascending), both arrays through the same network.
  for (int k = 2; k <= NPTS; k <<= 1) {
    for (int jj = k >> 1; jj > 0; jj >>= 1) {
      for (int i = tid; i < NPTS; i += nt) {
        const int ixj = i ^ jj;
        if (ixj > i) {
          const bool up = ((i & k) == 0);
          float a0 = sA[i], a1 = sA[ixj];
          if ((a0 > a1) == up) { sA[i] = a1; sA[ixj] = a0; }
          float b0 = sB[i], b1 = sB[ixj];
          if ((b0 > b1) == up) { sB[i] = b1; sB[ixj] = b0; }
        }
      }
      __syncthreads();
    }
  }

  float ordAcc = 0.0f;
  for (int i = tid; i < NPTS; i += nt) {
    const float d = sA[i] - sB[i];
    ordAcc = fmaf(d, d, ordAcc);
  }

  float v = crossAcc + ALPHA_F * ordAcc;
  __syncthreads();                        // sA reused as reduction scratch

  for (int off = 16; off > 0; off >>= 1) v += __shfl_xor(v, off, 32);
  if ((tid & 31) == 0) sA[tid >> 5] = v;
  __syncthreads();
  if (tid < 32) {
    float r = (tid < nt / 32) ? sA[tid] : 0.0f;
    for (int off = 16; off > 0; off >>= 1) r += __shfl_xor(r, off, 32);
    if (tid == 0) out[b] = r;
  }
}

// ---------------------------------------------------------------------------
extern "C" void kernel_launch(void* const* d_in, const int* in_sizes, int n_in,
                              void* d_out, int out_size, void* d_ws, size_t ws_size,
                              hipStream_t stream) {
  (void)in_sizes; (void)n_in; (void)out_size; (void)ws_size;

  const float* gts   = (const float*)d_in[0];
  const float* preds = (const float*)d_in[1];
  float* out = (float*)d_out;
  float* wsf = (float*)d_ws;   // 4 arrays of [4][4096] f32 = 256 KB

  const size_t lds = 4 * NPTS * sizeof(float);   // 64 KB dynamic LDS
  dim3 gridC(16, NBATCH, 2);   // 16 column groups x 4 batches x 2 matrices

  chamfer_min_kernel<false><<<gridC, 256, lds, stream>>>(gts, preds, wsf);  // cross
  chamfer_min_kernel<true ><<<gridC, 256, lds, stream>>>(gts, preds, wsf);  // self

  chamfer_finalize_kernel<<<NBATCH, 1024, 0, stream>>>(wsf, out);
}